// CombinedMarginLoss_2430951489682
// MI455X (gfx1250) — compile-verified
//
#include <hip/hip_runtime.h>
#include <cmath>

// ---------------------------------------------------------------------------
// CombinedMarginLoss (ArcFace, m1=1, m2=0.5, m3=0) forward, S=64.
//   out = S * cos(arccos(clip(x,-1,1)))            everywhere      == S*clip(x)
//   out[b, label[b]] = S * cos(arccos(clip(x)) + 0.5)   (label != -1)
// Memory-bound: 512MB in + 512MB out -> ~44us @ 23.3 TB/s. No matmul -> no
// WMMA; optimize the data-movement path instead: gfx1250 async-to-LDS
// streaming (ASYNCcnt double buffering) + non-temporal b128.
// ---------------------------------------------------------------------------

typedef float v4f __attribute__((ext_vector_type(4)));  // native vector: NT-ok
typedef int   v4i __attribute__((vector_size(16)));     // matches builtin param

#define TILE_F4 128   // v4f per tile = 4 b128 ops x 32 lanes (2 KB / tile)
#define OPS     4     // b128 async ops per stage per wave

#if defined(__HIP_DEVICE_COMPILE__) && __has_builtin(__builtin_amdgcn_global_load_async_to_lds_b128)
#define DEVICE_ASYNC 1
#else
#define DEVICE_ASYNC 0
#endif

#if defined(__HIP_DEVICE_COMPILE__) && __has_builtin(__builtin_amdgcn_s_wait_asynccnt)
#define WAIT_ASYNC(K)                                    \
  do {                                                   \
    asm volatile("" ::: "memory");                       \
    __builtin_amdgcn_s_wait_asynccnt(K);                 \
    asm volatile("" ::: "memory");                       \
  } while (0)
#else
#define WAIT_ASYNC(K) asm volatile("s_wait_asynccnt " #K ::: "memory")
#endif

#define AS1_V4I(p) ((__attribute__((address_space(1))) v4i*)(p))
#define AS3_V4I(p) ((__attribute__((address_space(3))) v4i*)(p))

__device__ __forceinline__ float xform(float x) {
  x = fminf(fmaxf(x, -1.0f), 1.0f);
  return x * 64.0f;
}

__device__ __forceinline__ v4f xform4(v4f v) {
  v.x = xform(v.x); v.y = xform(v.y); v.z = xform(v.z); v.w = xform(v.w);
  return v;
}

// Bulk pass: each wave streams whole 128-v4f tiles (fully coalesced 512B per
// b128 op, EXEC all-ones). Async path double-buffers through LDS; each wave
// only touches its own LDS slice -> no barriers needed.
__global__ void __launch_bounds__(256) bulk_tiles(const v4f* __restrict__ in4,
                                                  v4f* __restrict__ out4,
                                                  long long T) {
  const int lane  = threadIdx.x & 31;
  const int wslot = threadIdx.x >> 5;
  const long long wave   = (((long long)blockIdx.x * blockDim.x) + threadIdx.x) >> 5;
  const long long nwaves = ((long long)gridDim.x * blockDim.x) >> 5;
  long long t = wave;
  if (t >= T) return;  // wave-uniform

#if DEVICE_ASYNC
  __shared__ v4f lbuf[8][2][OPS * 32];  // 8 waves x 2 stages x 2KB = 32KB
  int p = 0;
  {
    const v4f* g = in4 + t * TILE_F4 + lane;
#pragma unroll
    for (int u = 0; u < OPS; ++u)
      __builtin_amdgcn_global_load_async_to_lds_b128(
          AS1_V4I(g + u * 32), AS3_V4I(&lbuf[wslot][p][u * 32 + lane]), 0, 0);
  }
  for (; t < T; t += nwaves) {
    const long long nxt = t + nwaves;
    if (nxt < T) {
      const v4f* g = in4 + nxt * TILE_F4 + lane;
#pragma unroll
      for (int u = 0; u < OPS; ++u)
        __builtin_amdgcn_global_load_async_to_lds_b128(
            AS1_V4I(g + u * 32), AS3_V4I(&lbuf[wslot][p ^ 1][u * 32 + lane]), 0, 0);
      WAIT_ASYNC(4);  // 4 newer ops in flight; current stage done (in-order)
    } else {
      WAIT_ASYNC(0);
    }
    v4f* o = out4 + t * TILE_F4 + lane;
#pragma unroll
    for (int u = 0; u < OPS; ++u) {
      v4f v = lbuf[wslot][p][u * 32 + lane];
      __builtin_nontemporal_store(xform4(v), o + u * 32);
    }
    p ^= 1;
  }
#else
  // Direct path: 4 independent NT b128 loads per tile + L2 prefetch of the
  // next tile this wave will touch.
  for (; t < T; t += nwaves) {
    const v4f* g = in4 + t * TILE_F4 + lane;
    v4f* o = out4 + t * TILE_F4 + lane;
    __builtin_prefetch(g + nwaves * TILE_F4, 0, 0);
    v4f v0 = __builtin_nontemporal_load(g);
    v4f v1 = __builtin_nontemporal_load(g + 32);
    v4f v2 = __builtin_nontemporal_load(g + 64);
    v4f v3 = __builtin_nontemporal_load(g + 96);
    __builtin_nontemporal_store(xform4(v0), o);
    __builtin_nontemporal_store(xform4(v1), o + 32);
    __builtin_nontemporal_store(xform4(v2), o + 64);
    __builtin_nontemporal_store(xform4(v3), o + 96);
  }
#endif
}

// Ragged tail (elements not covered by whole tiles). For N = 128e6 this is
// empty, but keep it generic.
__global__ void __launch_bounds__(256) tail_kernel(const float* __restrict__ in,
                                                   float* __restrict__ out,
                                                   long long start, long long n) {
  long long i = start + (long long)blockIdx.x * blockDim.x + threadIdx.x;
  const long long stride = (long long)gridDim.x * blockDim.x;
  for (; i < n; i += stride) out[i] = xform(in[i]);
}

// Margin fixup: one thread per row overwrites out[b, label[b]] with
// S*cos(acos(x)+0.5). f64 on 128 elements is free and sits well inside the
// reference's own f32 acos/cos rounding. Runs after bulk (same stream).
__global__ void fixup_kernel(const float* __restrict__ logits,
                             const int* __restrict__ labels,
                             float* __restrict__ out, int B, long long C) {
  const int b = blockIdx.x * blockDim.x + threadIdx.x;
  if (b >= B) return;
  const int lab = labels[b];
  if (lab < 0) return;  // invalid label -> reference leaves row unchanged
  const long long off = (long long)b * C + (long long)lab;
  const float x = fminf(fmaxf(logits[off], -1.0f), 1.0f);
  const double th = acos((double)x) + 0.5;
  out[off] = (float)(cos(th) * 64.0);
}

extern "C" void kernel_launch(void* const* d_in, const int* in_sizes, int n_in,
                              void* d_out, int out_size, void* d_ws, size_t ws_size,
                              hipStream_t stream) {
  (void)d_ws; (void)ws_size; (void)out_size;
  const float* logits = (const float*)d_in[0];
  const int*   labels = (const int*)d_in[1];
  float* out = (float*)d_out;

  const long long N  = (long long)in_sizes[0];
  const int B        = (n_in > 1) ? in_sizes[1] : 0;
  const long long C  = (B > 0) ? (N / (long long)B) : 1;
  const long long n4 = N >> 2;
  const long long T  = n4 / TILE_F4;   // whole tiles

  if (T > 0) {
    long long blocks = (T + 7) / 8;    // 8 waves per 256-thread block
    if (blocks > 4096) blocks = 4096;  // persistent-ish: ~8 tiles/wave at N=128e6
    bulk_tiles<<<(int)blocks, 256, 0, stream>>>((const v4f*)logits,
                                                (v4f*)out, T);
  }
  const long long start = T * (long long)(TILE_F4 * 4);
  if (start < N) {
    long long rem = N - start;
    long long blocks = (rem + 255) / 256;
    if (blocks > 1024) blocks = 1024;
    tail_kernel<<<(int)blocks, 256, 0, stream>>>(logits, out, start, N);
  }
  if (B > 0) {
    fixup_kernel<<<(B + 255) / 256, 256, 0, stream>>>(logits, labels, out, B, C);
  }
}